// CotangentConv_50448685859291
// MI455X (gfx1250) — compile-verified
//
#include <hip/hip_runtime.h>
#include <hip/hip_bf16.h>
#include <math.h>

#define N 3072
#define CIN 128
#define COUT 128
#define EH 32
#define RES_SCALE 0.1f
#define RED_BLOCKS 1024
#define RED_THREADS 256
#define CAP 1024

typedef float v2f __attribute__((ext_vector_type(2)));
typedef float v8f __attribute__((ext_vector_type(8)));

// ---------------------------------------------------------------------------
// Edge MLP: delta = b2 + sum_k silu(feat . w1[:,k] + b1[k]) * w2[k]
// w1 is (4, EH) row-major; feat = [dx, dy, dz, dist]
// ---------------------------------------------------------------------------
__device__ __forceinline__ float edge_delta(float dx, float dy, float dz, float dist,
                                            const float* sw1, const float* sb1,
                                            const float* sw2, float b2v) {
    float d = b2v;
#pragma unroll
    for (int k = 0; k < EH; ++k) {
        float h = fmaf(dx, sw1[k],
                  fmaf(dy, sw1[EH + k],
                  fmaf(dz, sw1[2 * EH + k],
                  fmaf(dist, sw1[3 * EH + k], sb1[k]))));
        // silu(h) = h * sigmoid(h)
        h = h / (1.0f + expf(-h));
        d = fmaf(h, sw2[k], d);
    }
    return d;
}

// ---------------------------------------------------------------------------
// Kernel 1: Y = x @ weight  (3072x128 @ 128x128, all f32) via V_WMMA_F32_16X16X4_F32
// One wave per 16x16 output tile; 8 waves/block cover all 128 columns.
// ---------------------------------------------------------------------------
__global__ __launch_bounds__(256) void gemm_y_wmma(const float* __restrict__ x,
                                                   const float* __restrict__ w,
                                                   float* __restrict__ Y) {
    const int lane = threadIdx.x & 31;
    const int wv   = threadIdx.x >> 5;
    const int row0 = blockIdx.x * 16;
    const int col0 = wv * 16;
    const int m    = lane & 15;           // row (A) / col (B) within tile
    const int kh   = (lane >> 4) << 1;    // 0 for lanes 0-15, 2 for lanes 16-31

    v8f c = {0.f, 0.f, 0.f, 0.f, 0.f, 0.f, 0.f, 0.f};
    const float* xr = x + (size_t)(row0 + m) * CIN;

#pragma unroll 4
    for (int k0 = 0; k0 < CIN; k0 += 4) {
        v2f a, b;
        // A 16x4: VGPR0 = A[m][kh], VGPR1 = A[m][kh+1]
        a.x = xr[k0 + kh];
        a.y = xr[k0 + kh + 1];
        // B 4x16: VGPR0 = B[kh][n], VGPR1 = B[kh+1][n]
        b.x = w[(size_t)(k0 + kh) * COUT + col0 + m];
        b.y = w[(size_t)(k0 + kh + 1) * COUT + col0 + m];
        c = __builtin_amdgcn_wmma_f32_16x16x4_f32(false, a, false, b,
                                                  (short)0, c, false, false);
    }
#pragma unroll
    for (int r = 0; r < 8; ++r) {
        int mm = r + ((lane >> 4) << 3);  // VGPR r: M = r (lanes 0-15), r+8 (lanes 16-31)
        Y[(size_t)(row0 + mm) * COUT + col0 + m] = c[r];
    }
}

// ---------------------------------------------------------------------------
// Kernel 2: colsum[c] = sum_i Y[i][c]  (rank-1 term from the mean subtraction)
// Single block, deterministic serial-per-column accumulation (1.5 MB read).
// ---------------------------------------------------------------------------
__global__ __launch_bounds__(128) void colsum_kernel(const float* __restrict__ Y,
                                                     float* __restrict__ colsum) {
    const int c = threadIdx.x;
    float s = 0.f;
    for (int i = 0; i < N; ++i) s += Y[(size_t)i * COUT + c];
    colsum[c] = s;
}

// ---------------------------------------------------------------------------
// Kernel 3: one full scan of adj. Per block partial sums of:
//   nedges = sum(adj), sum_d2 = sum(d2*adj), sum_delta = sum(delta on edges)
// Deterministic: fixed grid, in-block tree reduce, partials to fixed slots.
// ---------------------------------------------------------------------------
__global__ __launch_bounds__(RED_THREADS) void edge_reduce_kernel(
        const float* __restrict__ adj, const float* __restrict__ pos,
        const float* __restrict__ w1, const float* __restrict__ b1,
        const float* __restrict__ w2, const float* __restrict__ b2,
        float* __restrict__ partials) {
    __shared__ float sw1[4 * EH], sb1[EH], sw2[EH];
    __shared__ float sb2s;
    __shared__ float r0[RED_THREADS], r1[RED_THREADS], r2[RED_THREADS];
    const int tid = threadIdx.x;
    if (tid < 4 * EH) sw1[tid] = w1[tid];
    if (tid < EH) { sb1[tid] = b1[tid]; sw2[tid] = w2[tid]; }
    if (tid == 0) sb2s = b2[0];
    __syncthreads();

    float ne = 0.f, sd2 = 0.f, sdel = 0.f;
    const size_t total  = (size_t)N * N;
    const size_t stride = (size_t)gridDim.x * blockDim.x;
    for (size_t idx = (size_t)blockIdx.x * blockDim.x + tid; idx < total; idx += stride) {
        float a = adj[idx];
        if (a > 0.f) {
            int i = (int)(idx / N), j = (int)(idx % N);
            float dx = pos[3 * i]     - pos[3 * j];
            float dy = pos[3 * i + 1] - pos[3 * j + 1];
            float dz = pos[3 * i + 2] - pos[3 * j + 2];
            float d2 = dx * dx + dy * dy + dz * dz;
            ne  += a;
            sd2 += d2 * a;
            float dist = fmaxf(sqrtf(fmaxf(d2, 1e-16f)), 1e-8f);
            sdel += edge_delta(dx, dy, dz, dist, sw1, sb1, sw2, sb2s);
        }
    }
    r0[tid] = ne; r1[tid] = sd2; r2[tid] = sdel;
    __syncthreads();
    for (int s = RED_THREADS / 2; s > 0; s >>= 1) {
        if (tid < s) { r0[tid] += r0[tid + s]; r1[tid] += r1[tid + s]; r2[tid] += r2[tid + s]; }
        __syncthreads();
    }
    if (tid == 0) {
        partials[blockIdx.x]                  = r0[0];
        partials[RED_BLOCKS + blockIdx.x]     = r1[0];
        partials[2 * RED_BLOCKS + blockIdx.x] = r2[0];
    }
}

// ---------------------------------------------------------------------------
// Kernel 4: deterministic final reduce of the per-block partials -> scalars[3]
// ---------------------------------------------------------------------------
__global__ __launch_bounds__(256) void finalize_kernel(const float* __restrict__ partials,
                                                       float* __restrict__ scalars) {
    __shared__ float r[256];
    const int tid = threadIdx.x;
    for (int q = 0; q < 3; ++q) {
        float s = 0.f;
        for (int b = tid; b < RED_BLOCKS; b += 256) s += partials[q * RED_BLOCKS + b];
        r[tid] = s;
        __syncthreads();
        for (int st = 128; st > 0; st >>= 1) {
            if (tid < st) r[tid] += r[tid + st];
            __syncthreads();
        }
        if (tid == 0) scalars[q] = r[0];
        __syncthreads();
    }
}

// ---------------------------------------------------------------------------
// Kernel 5: one block per output row i.
//  - deterministic neighbor-list build (count -> prefix -> fill) into LDS
//  - D = rowsum(W); m_ij = w_ij/D + 0.1*delta_ij
//  - out[i,c] = sum_j m_ij * Y[j,c] - 0.1*mean(delta)*colsum[c] + bias[c]
// ---------------------------------------------------------------------------
__global__ __launch_bounds__(128) void out_kernel(
        const float* __restrict__ adj, const float* __restrict__ pos,
        const float* __restrict__ Y, const float* __restrict__ colsum,
        const float* __restrict__ bias,
        const float* __restrict__ w1, const float* __restrict__ b1,
        const float* __restrict__ w2, const float* __restrict__ b2,
        const float* __restrict__ scalars, float* __restrict__ out) {
    __shared__ float sw1[4 * EH], sb1[EH], sw2[EH];
    __shared__ float sb2s;
    __shared__ int   s_off[129];
    __shared__ int   s_j[CAP];
    __shared__ float s_w[CAP], s_d[CAP];
    __shared__ float sD;

    const int tid = threadIdx.x;
    const int i   = blockIdx.x;
    if (tid < 4 * EH) sw1[tid] = w1[tid];
    if (tid < EH) { sb1[tid] = b1[tid]; sw2[tid] = w2[tid]; }
    if (tid == 0) sb2s = b2[0];
    __syncthreads();

    const float ns        = scalars[0];
    const float sum_d2    = scalars[1];
    const float sum_delta = scalars[2];
    const float t        = fmaxf(sum_d2 / fmaxf(ns, 1.0f), 1e-8f);
    const float inv4t    = 0.25f / t;
    const float meanterm = RES_SCALE * (sum_delta / ((float)N * (float)N));

    const float pix = pos[3 * i], piy = pos[3 * i + 1], piz = pos[3 * i + 2];
    const float* arow = adj + (size_t)i * N;

    // pass A: per-thread edge count (deterministic assignment j = tid + 128*k)
    int cnt_t = 0;
    for (int j = tid; j < N; j += 128)
        if (arow[j] > 0.f) cnt_t++;
    s_off[tid] = cnt_t;
    __syncthreads();
    if (tid == 0) {
        int acc = 0;
        for (int k = 0; k < 128; ++k) { int c = s_off[k]; s_off[k] = acc; acc += c; }
        s_off[128] = acc;
    }
    __syncthreads();
    int off   = s_off[tid];
    int total = s_off[128];
    if (total > CAP) total = CAP;

    // pass B: fill neighbor list with (j, w_ij, delta_ij)
    for (int j = tid; j < N; j += 128) {
        float a = arow[j];
        if (a > 0.f) {
            float dx = pix - pos[3 * j];
            float dy = piy - pos[3 * j + 1];
            float dz = piz - pos[3 * j + 2];
            float d2 = dx * dx + dy * dy + dz * dz;
            float wgt  = expf(-d2 * inv4t) * a;   // adj diag is 0 -> (1-eye) implicit
            float dist = fmaxf(sqrtf(fmaxf(d2, 1e-16f)), 1e-8f);
            float del  = edge_delta(dx, dy, dz, dist, sw1, sb1, sw2, sb2s);
            if (off < CAP) { s_j[off] = j; s_w[off] = wgt; s_d[off] = del; }
            off++;
        }
    }
    __syncthreads();
    if (tid == 0) {
        float D = 0.f;
        for (int e = 0; e < total; ++e) D += s_w[e];
        sD = fmaxf(D, 1e-8f);
    }
    __syncthreads();

    const float invD = 1.0f / sD;
    float acc = 0.f;
    for (int e = 0; e < total; ++e) {
        float m = fmaf(RES_SCALE, s_d[e], s_w[e] * invD);
        acc = fmaf(m, Y[(size_t)s_j[e] * COUT + tid], acc);
    }
    out[(size_t)i * COUT + tid] = acc - meanterm * colsum[tid] + bias[tid];
}

// ---------------------------------------------------------------------------
// kernel_launch: ws layout (floats):
//   [0..2]   scalars (nedges, sum_d2adj, sum_delta)
//   [64..191] colsum (128)
//   [256 .. 256+N*COUT)             Y
//   [256+N*COUT .. +3*RED_BLOCKS)   reduction partials
// ---------------------------------------------------------------------------
extern "C" void kernel_launch(void* const* d_in, const int* in_sizes, int n_in,
                              void* d_out, int out_size, void* d_ws, size_t ws_size,
                              hipStream_t stream) {
    const float* x    = (const float*)d_in[0];
    const float* pos  = (const float*)d_in[1];
    const float* adj  = (const float*)d_in[2];
    const float* wgt  = (const float*)d_in[3];
    const float* bias = (const float*)d_in[4];
    const float* w1   = (const float*)d_in[5];
    const float* b1   = (const float*)d_in[6];
    const float* w2   = (const float*)d_in[7];
    const float* b2   = (const float*)d_in[8];
    float* out = (float*)d_out;

    float* wsf      = (float*)d_ws;
    float* scalars  = wsf;
    float* colsum   = wsf + 64;
    float* Y        = wsf + 256;
    float* partials = wsf + 256 + (size_t)N * COUT;

    gemm_y_wmma<<<N / 16, 256, 0, stream>>>(x, wgt, Y);
    colsum_kernel<<<1, 128, 0, stream>>>(Y, colsum);
    edge_reduce_kernel<<<RED_BLOCKS, RED_THREADS, 0, stream>>>(adj, pos, w1, b1, w2, b2, partials);
    finalize_kernel<<<1, 256, 0, stream>>>(partials, scalars);
    out_kernel<<<N, 128, 0, stream>>>(adj, pos, Y, colsum, bias, w1, b1, w2, b2, scalars, out);
}